// Point2Voxel_69337952027016
// MI455X (gfx1250) — compile-verified
//
#include <hip/hip_runtime.h>
#include <hip/hip_bf16.h>
#include <stdint.h>

#define NB    8192          // max linear voxel bins per batch (<= 20^3 = 8000 for [0,1) input)
#define TILE  1024          // points per async tile (1024 * 12B = 12 KB)
#define BDIM  256           // 8 waves (wave32)
#define RADIUS_F 0.05f

// ---------------- K0: init min/max + bin_map ----------------
__global__ void init_kernel(int* kmin, int* kmax, int* binmap) {
    int t = threadIdx.x;
    if (t < 24) { kmin[t] = 0x7fffffff; kmax[t] = (int)0x80000000; }
    if (t < 81) {
        int x = t / 3, d = t % 3;
        int dig = (d == 0) ? (x / 9) : (d == 1) ? ((x / 3) % 3) : (x % 3);
        binmap[t] = dig - 1;
    }
}

// ---------------- K1: keys = floor(p/0.05), per-batch min/max ----------------
// wave32 butterfly reduction (5 shfl_xor steps), then 1 set of global atomics per wave.
__global__ __launch_bounds__(BDIM) void keys_minmax_kernel(
    const float* __restrict__ pts, int* __restrict__ keys,
    int* kmin, int* kmax, int N) {
    int b = blockIdx.y;
    long j = (long)blockIdx.x * BDIM + threadIdx.x;

    int k0, k1, k2;
    int mn0 = 0x7fffffff, mn1 = 0x7fffffff, mn2 = 0x7fffffff;
    int mx0 = (int)0x80000000, mx1 = mx0, mx2 = mx0;
    if (j < N) {
        const float* p = pts + ((long)b * N + j) * 3;
        float q0 = p[0], q1 = p[1], q2 = p[2];                 // global_load_b96
        k0 = (int)floorf(q0 / RADIUS_F);                       // match reference: divide then floor
        k1 = (int)floorf(q1 / RADIUS_F);
        k2 = (int)floorf(q2 / RADIUS_F);
        int* kp = keys + ((long)b * N + j) * 3;
        kp[0] = k0; kp[1] = k1; kp[2] = k2;                    // global_store_b96
        mn0 = mx0 = k0; mn1 = mx1 = k1; mn2 = mx2 = k2;
    }
    // wave32 reduction
    #pragma unroll
    for (int m = 16; m >= 1; m >>= 1) {
        mn0 = min(mn0, __shfl_xor(mn0, m, 32));
        mn1 = min(mn1, __shfl_xor(mn1, m, 32));
        mn2 = min(mn2, __shfl_xor(mn2, m, 32));
        mx0 = max(mx0, __shfl_xor(mx0, m, 32));
        mx1 = max(mx1, __shfl_xor(mx1, m, 32));
        mx2 = max(mx2, __shfl_xor(mx2, m, 32));
    }
    if ((threadIdx.x & 31) == 0) {
        atomicMin(&kmin[b * 3 + 0], mn0);
        atomicMin(&kmin[b * 3 + 1], mn1);
        atomicMin(&kmin[b * 3 + 2], mn2);
        atomicMax(&kmax[b * 3 + 0], mx0);
        atomicMax(&kmax[b * 3 + 1], mx1);
        atomicMax(&kmax[b * 3 + 2], mx2);
    }
}

// ---------------- K2: row-major strides + bin count per batch ----------------
__global__ void strides_kernel(const int* kmin, const int* kmax, int* sdat) {
    int b = threadIdx.x;
    if (b >= 8) return;
    int r0 = kmax[b*3+0] - kmin[b*3+0] + 1;
    int r1 = kmax[b*3+1] - kmin[b*3+1] + 1;
    int r2 = kmax[b*3+2] - kmin[b*3+2] + 1;
    sdat[b*4+0] = r1 * r2;       // stride for dim0
    sdat[b*4+1] = r2;            // stride for dim1
    sdat[b*4+2] = r0 * r1 * r2;  // nbins
    sdat[b*4+3] = 0;
}

// ---------------- K3: async-LDS-fed privatized histogram ----------------
__global__ __launch_bounds__(BDIM) void hist_kernel(
    const int* __restrict__ keys, const int* kmin, const int* sdat,
    int* __restrict__ ghist, int N, int tilesPerBlock) {
    __shared__ int hist[NB];             // 32 KB
    __shared__ int kbuf[2][TILE * 3];    // 2 x 12 KB double buffer
    const int b = blockIdx.y;
    const int t = threadIdx.x;

    for (int i = t; i < NB; i += BDIM) hist[i] = 0;
    const int m0 = kmin[b*3+0], m1 = kmin[b*3+1], m2 = kmin[b*3+2];
    const int s0 = sdat[b*4+0], s1 = sdat[b*4+1];
    __syncthreads();

    const int* kb = keys + (long)b * N * 3;   // uniform -> SGPR pair for saddr
    const int numTiles = (N + TILE - 1) / TILE;
    const int tile0 = blockIdx.x * tilesPerBlock;
    const int tileEnd = min(tile0 + tilesPerBlock, numTiles);

    unsigned ldsb0 = (unsigned)(uintptr_t)&kbuf[0][0];  // LDS byte offset (low 32 bits of aperture addr)
    unsigned ldsb1 = (unsigned)(uintptr_t)&kbuf[1][0];

    // CDNA5 async DMA: global -> LDS, tracked by ASYNCcnt. 3 x b128 per thread per tile.
    auto issue_tile = [&](int tile, int buf) {
        unsigned gof = (unsigned)tile * (TILE * 12u);
        unsigned base = buf ? ldsb1 : ldsb0;
        #pragma unroll
        for (int r = 0; r < 3; ++r) {
            unsigned off = (t + r * BDIM) * 16u;
            unsigned la = base + off;
            unsigned va = gof + off;
            asm volatile("global_load_async_to_lds_b128 %0, %1, %2"
                         :: "v"(la), "v"(va), "s"(kb) : "memory");
        }
    };

    if (tile0 < tileEnd) issue_tile(tile0, tile0 & 1);
    for (int tile = tile0; tile < tileEnd; ++tile) {
        if (tile + 1 < tileEnd) {
            issue_tile(tile + 1, (tile + 1) & 1);
            asm volatile("s_wait_asynccnt 0x3" ::: "memory");  // in-order: current tile's 3 are done
        } else {
            asm volatile("s_wait_asynccnt 0x0" ::: "memory");
        }
        __syncthreads();
        const int* kk = kbuf[tile & 1];
        const int jbase = tile * TILE;
        #pragma unroll
        for (int r = 0; r < TILE / BDIM; ++r) {
            int idx = t + r * BDIM;
            int j = jbase + idx;
            if (j < N) {
                int k0 = kk[idx*3+0], k1 = kk[idx*3+1], k2 = kk[idx*3+2];
                int lin = (k0 - m0) * s0 + (k1 - m1) * s1 + (k2 - m2);
                if (lin >= 0 && lin < NB) atomicAdd(&hist[lin], 1);   // ds_add_u32
            }
        }
        __syncthreads();  // protect buffer (tile&1) before it is re-filled at tile+2
    }

    // sparse merge to the per-batch global histogram
    for (int i = t; i < NB; i += BDIM) {
        int c = hist[i];
        if (c) atomicAdd(&ghist[(long)b * NB + i], c);
    }
}

// ---------------- K4: exclusive scan of occupancy -> dense voxel ranks ----------------
__global__ __launch_bounds__(BDIM) void scan_kernel(
    const int* __restrict__ ghist, int* __restrict__ grank, const int* sdat) {
    const int b = blockIdx.x;
    const int* h = ghist + (long)b * NB;
    int* r = grank + (long)b * NB;
    int nbins = sdat[b*4+2];
    if (nbins > NB) nbins = NB;
    __shared__ int part[BDIM];
    const int t = threadIdx.x;
    const int per = NB / BDIM;   // 32 bins/thread
    const int base = t * per;
    int cnt = 0;
    for (int i = 0; i < per; ++i) {
        int bin = base + i;
        if (bin < nbins && h[bin] != 0) cnt++;
    }
    part[t] = cnt;
    __syncthreads();
    for (int off = 1; off < BDIM; off <<= 1) {   // Hillis-Steele inclusive scan
        int v = (t >= off) ? part[t - off] : 0;
        __syncthreads();
        part[t] += v;
        __syncthreads();
    }
    int run = part[t] - cnt;                     // exclusive base for this chunk
    for (int i = 0; i < per; ++i) {
        int bin = base + i;
        if (bin < nbins) {
            r[bin] = run;
            if (h[bin] != 0) run++;
        }
    }
}

// ---------------- K5: gather voxel_id + counts per point ----------------
__global__ __launch_bounds__(BDIM) void gather_kernel(
    const int* __restrict__ keys, const int* kmin, const int* sdat,
    const int* __restrict__ ghist, const int* __restrict__ grank,
    int* __restrict__ vox, int* __restrict__ cnt, int N) {
    int b = blockIdx.y;
    long j = (long)blockIdx.x * BDIM + threadIdx.x;
    if (j >= N) return;
    const int* k = keys + ((long)b * N + j) * 3;
    int k0 = k[0], k1 = k[1], k2 = k[2];                       // global_load_b96
    int m0 = kmin[b*3+0], m1 = kmin[b*3+1], m2 = kmin[b*3+2];
    int s0 = sdat[b*4+0], s1 = sdat[b*4+1];
    int lin = (k0 - m0) * s0 + (k1 - m1) * s1 + (k2 - m2);
    if (lin < 0) lin = 0;
    if (lin >= NB) lin = NB - 1;
    vox[(long)b * N + j] = grank[(long)b * NB + lin];
    cnt[(long)b * N + j] = ghist[(long)b * NB + lin];
}

extern "C" void kernel_launch(void* const* d_in, const int* in_sizes, int n_in,
                              void* d_out, int out_size, void* d_ws, size_t ws_size,
                              hipStream_t stream) {
    const float* pts = (const float*)d_in[0];
    const int B = 8, D = 3;
    const int N = in_sizes[0] / (B * D);

    int* out = (int*)d_out;
    int* keys = out;                               // [B, N, 3]
    long keysCnt = (long)B * N * 3;
    int* vox = out + keysCnt;                      // [B, N]
    int* cnt = vox + (long)B * N;                  // [B, N]
    int* bin = cnt + (long)B * N;                  // [27, 3]

    int* ws    = (int*)d_ws;
    int* kmin  = ws;                               // 24 ints
    int* kmax  = ws + 24;                          // 24 ints
    int* sdat  = ws + 48;                          // 32 ints (s0,s1,nbins,pad)*8
    int* ghist = ws + 128;                         // B*NB
    int* grank = ghist + (long)B * NB;             // B*NB

    hipMemsetAsync(ghist, 0, (size_t)B * NB * sizeof(int), stream);
    init_kernel<<<1, 128, 0, stream>>>(kmin, kmax, bin);

    dim3 gPts((N + BDIM - 1) / BDIM, B);
    keys_minmax_kernel<<<gPts, BDIM, 0, stream>>>(pts, keys, kmin, kmax, N);
    strides_kernel<<<1, 32, 0, stream>>>(kmin, kmax, sdat);

    const int numTiles = (N + TILE - 1) / TILE;
    const int blocksPerBatch = 32;
    const int tilesPerBlock = (numTiles + blocksPerBatch - 1) / blocksPerBatch;
    hist_kernel<<<dim3(blocksPerBatch, B), BDIM, 0, stream>>>(keys, kmin, sdat, ghist, N, tilesPerBlock);

    scan_kernel<<<dim3(B), BDIM, 0, stream>>>(ghist, grank, sdat);
    gather_kernel<<<gPts, BDIM, 0, stream>>>(keys, kmin, sdat, ghist, grank, vox, cnt, N);
}